// ByteMemory_24043226923976
// MI455X (gfx1250) — compile-verified
//
#include <hip/hip_runtime.h>
#include <cstdint>

// ---------------------------------------------------------------------------
// ByteMemory n-gram hash gather for MI455X (gfx1250, wave32)
//
//   out[b, w, :] = memory_table[ fnv3(input_bytes[b, w:w+3]) % 1e6 , : ]
//
// Bandwidth-bound 512B-row gather. CDNA5 async DMA path:
//   HBM --global_load_async_to_lds_b128--> LDS
//       --global_store_async_from_lds_b128--> HBM
// Payload never touches VGPRs; ASYNCcnt pipelines 8 row transfers per wave.
// Hash indices are computed on the scalar path (uniform per wave).
// ---------------------------------------------------------------------------

#define CAPACITY   1000000u
#define MEM_DIM    128
#define ROW_BYTES  (MEM_DIM * 4)           /* 512 B per table row  */
#define HASH_SEED  0x12345678u
#define FNV_PRIME  16777619u

#define BATCH      32
#define SEQ_LEN    8192
#define OUT_LEN    8190                    /* SEQ_LEN - 3 + 1      */
#define N_WINDOWS  (BATCH * OUT_LEN)       /* 262080               */

#define WAVES_PER_BLOCK 8
#define WIN_PER_WAVE    8
#define BLOCK_THREADS   (WAVES_PER_BLOCK * 32)
#define N_BLOCKS        (N_WINDOWS / (WAVES_PER_BLOCK * WIN_PER_WAVE)) /* 4095 */

__global__ __launch_bounds__(BLOCK_THREADS) void byte_memory_gather_async(
    const int*   __restrict__ input_bytes,   // [32, 8192] int32 in [0,256)
    const float* __restrict__ table,         // [1e6, 128] f32
    float*       __restrict__ out)           // [32, 8190, 128] f32
{
    // 8 waves * 8 rows * 512 B = 32 KB staging (of 320 KB WGP LDS)
    __shared__ char stage[WAVES_PER_BLOCK * WIN_PER_WAVE * ROW_BYTES];

    const unsigned lane = threadIdx.x & 31u;
    // Force wave-uniform so the whole hash path scalarizes (SMEM + SALU).
    const unsigned wave =
        (unsigned)__builtin_amdgcn_readfirstlane((int)(threadIdx.x >> 5));
    const unsigned gw = blockIdx.x * WAVES_PER_BLOCK + wave;   // global wave id
    const unsigned w0 = gw * WIN_PER_WAVE;                     // first window

    // Wave-private LDS staging base; low 32 bits of a flat LDS-aperture
    // pointer are the wave-relative LDS byte address (ISA 10.2).
    const unsigned lds_wave =
        (unsigned)(uintptr_t)&stage[0]
        + wave * (WIN_PER_WAVE * ROW_BYTES)
        + lane * 16u;

    // ---- scalar phase: 8 rolling hashes -> 8 table row indices ------------
    unsigned row[WIN_PER_WAVE];
#pragma unroll
    for (int k = 0; k < WIN_PER_WAVE; ++k) {
        const unsigned w   = w0 + (unsigned)k;
        const unsigned b   = w / OUT_LEN;          // const-div -> mulhi (SALU)
        const unsigned pos = w - b * OUT_LEN;
        const int* p = input_bytes + (size_t)b * SEQ_LEN + pos;
        unsigned h = HASH_SEED;
        h = (h * FNV_PRIME) ^ (unsigned)p[0];
        h = (h * FNV_PRIME) ^ (unsigned)p[1];
        h = (h * FNV_PRIME) ^ (unsigned)p[2];
        row[k] = h % CAPACITY;
    }

    // ---- async gather: HBM -> LDS, one b128 per lane (512 B per row) ------
#pragma unroll
    for (int k = 0; k < WIN_PER_WAVE; ++k) {
        const uint64_t ga = (uint64_t)(uintptr_t)
            ((const char*)table + (size_t)row[k] * ROW_BYTES + lane * 16u);
        const unsigned la = lds_wave + (unsigned)k * ROW_BYTES;
        asm volatile("global_load_async_to_lds_b128 %0, %1, off"
                     :: "v"(la), "v"(ga) : "memory");
    }

    // All 8 row transfers must land in LDS before we stream them out.
    asm volatile("s_wait_asynccnt 0x0" ::: "memory");

    // ---- async stream-out: LDS -> HBM, non-temporal (protect L2) ----------
#pragma unroll
    for (int k = 0; k < WIN_PER_WAVE; ++k) {
        const uint64_t ga = (uint64_t)(uintptr_t)
            ((char*)out + (size_t)(w0 + (unsigned)k) * ROW_BYTES + lane * 16u);
        const unsigned la = lds_wave + (unsigned)k * ROW_BYTES;
        asm volatile("global_store_async_from_lds_b128 %0, %1, off th:TH_STORE_NT"
                     :: "v"(ga), "v"(la) : "memory");
    }
    // s_endpgm performs an implicit wait-idle: async stores complete before
    // the wave retires, so no explicit trailing s_wait_asynccnt is needed.
}

extern "C" void kernel_launch(void* const* d_in, const int* in_sizes, int n_in,
                              void* d_out, int out_size, void* d_ws, size_t ws_size,
                              hipStream_t stream)
{
    (void)in_sizes; (void)n_in; (void)out_size; (void)d_ws; (void)ws_size;
    const int*   input_bytes = (const int*)  d_in[0];  // [32, 8192] int32
    const float* table       = (const float*)d_in[1];  // [1e6, 128] f32
    float*       out         = (float*)      d_out;    // [32, 8190, 128] f32

    byte_memory_gather_async<<<dim3(N_BLOCKS), dim3(BLOCK_THREADS), 0, stream>>>(
        input_bytes, table, out);
}